// EdgeAwareAttention_56564719288944
// MI455X (gfx1250) — compile-verified
//
#include <hip/hip_runtime.h>
#include <hip/hip_bf16.h>

typedef float f32x2 __attribute__((ext_vector_type(2)));
typedef float f32x8 __attribute__((ext_vector_type(8)));

#define NODE_DIM 128
#define EDGE_DIM 8
#define NUM_HEADS 8

// ---------------------------------------------------------------------------
// K0: zero the output accumulator (d_out is poisoned by the harness)
// ---------------------------------------------------------------------------
__global__ void ea_zero_kernel(float4* __restrict__ out, int n4) {
    int t = blockIdx.x * blockDim.x + threadIdx.x;
    if (t < n4) out[t] = make_float4(0.f, 0.f, 0.f, 0.f);
}

// ---------------------------------------------------------------------------
// K1: x_proj = x @ W_node^T + b_node   via V_WMMA_F32_16X16X4_F32
//   One wave computes one 16x16 output tile; 8 waves per block cover the
//   8 N-tiles (NODE_DIM=128) of one 16-row M-tile.
//   A (16x4, M x K) fp32 layout: lanes 0-15 -> M=lane,  VGPR{0,1}=K{0,1}
//                                lanes 16-31 -> M=lane-16, VGPR{0,1}=K{2,3}
//   B (4x16, K x N) fp32 layout: lanes 0-15 -> N=lane,  VGPR{0,1}=K{0,1}
//                                lanes 16-31 -> N=lane-16, VGPR{0,1}=K{2,3}
//   B[k][n] = W_node[n][k]  -> contiguous float2 reads of W_node rows.
//   C/D (16x16) layout: VGPR r, lanes 0-15 -> (M=r,   N=lane)
//                               lanes 16-31 -> (M=r+8, N=lane-16)
// ---------------------------------------------------------------------------
__global__ void ea_node_proj_wmma(const float* __restrict__ x,
                                  const float* __restrict__ W_node,  // [128,128]
                                  const float* __restrict__ b_node,  // [128]
                                  float* __restrict__ x_proj,
                                  int nrows) {
    const int wave  = threadIdx.x >> 5;   // 0..7 -> N tile
    const int lane  = threadIdx.x & 31;
    const int half  = lane >> 4;          // which half-wave
    const int l16   = lane & 15;
    const int row0  = blockIdx.x * 16;    // M tile base

    const int col   = wave * 16 + l16;                 // output column (N) of this lane
    int arow        = row0 + l16;                      // A-matrix row (M) of this lane
    if (arow >= nrows) arow = nrows - 1;               // safe clamp (reads only)

    // Accumulator pre-loaded with the bias: C[m][n] = b_node[n], n depends
    // only on the lane, so all 8 VGPRs get the same per-lane value.
    const float bias = b_node[col];
    f32x8 acc;
#pragma unroll
    for (int i = 0; i < 8; ++i) acc[i] = bias;

    const float* __restrict__ xrow = x      + (size_t)arow * NODE_DIM;
    const float* __restrict__ wrow = W_node + (size_t)col  * NODE_DIM;

#pragma unroll
    for (int k0 = 0; k0 < NODE_DIM; k0 += 4) {
        const int ka = k0 + half * 2;
        f32x2 a = *(const f32x2*)(xrow + ka);          // x[arow, ka..ka+1]
        f32x2 b = *(const f32x2*)(wrow + ka);          // W_node[col, ka..ka+1]
        // 8 args: (neg_a, A, neg_b, B, c_mod, C, reuse_a, reuse_b)
        acc = __builtin_amdgcn_wmma_f32_16x16x4_f32(
            false, a, false, b, (short)0, acc, false, false);
    }

    // Store: acc[r] -> (M = row0 + r + half*8, N = col); coalesced per VGPR.
#pragma unroll
    for (int r = 0; r < 8; ++r) {
        const int m = row0 + r + half * 8;
        if (m < nrows) x_proj[(size_t)m * NODE_DIM + col] = acc[r];
    }
}

// ---------------------------------------------------------------------------
// K2: fused gate + gather + scatter-add.  One wave per edge; lane l owns
//     output elements [4l, 4l+3] (all inside head l>>2).
// ---------------------------------------------------------------------------
__global__ void ea_edge_scatter(const float* __restrict__ x_proj,
                                const int*   __restrict__ edge_index, // [2,E]
                                const float* __restrict__ edge_attr,  // [E,8]
                                const float* __restrict__ W_edge,     // [8,8]
                                const float* __restrict__ b_edge,     // [8]
                                float* __restrict__ out,
                                int E_) {
    const long t    = (long)blockIdx.x * blockDim.x + threadIdx.x;
    const int  e    = (int)(t >> 5);
    const int  lane = (int)(t & 31);
    if (e >= E_) return;

    const int src  = edge_index[e];        // row 0 of edge_index
    const int tgt  = edge_index[E_ + e];   // row 1 of edge_index
    const int head = lane >> 2;

    // gate = sigmoid(edge_attr[e] . W_edge[head] + b_edge[head])
    const float* __restrict__ ea = edge_attr + (size_t)e * EDGE_DIM;  // 32B wave-broadcast
    const float* __restrict__ wr = W_edge + head * EDGE_DIM;          // 256B table, cached
    float z = b_edge[head];
#pragma unroll
    for (int k = 0; k < EDGE_DIM; ++k) z = fmaf(ea[k], wr[k], z);
    const float gate = 1.0f / (1.0f + expf(-z));

    // gather one full 512B row per wave (global_load_b128 per lane)
    const float4 v = *(const float4*)(x_proj + (size_t)src * NODE_DIM + lane * 4);

    float* __restrict__ dst = out + (size_t)tgt * NODE_DIM + lane * 4;
    unsafeAtomicAdd(dst + 0, v.x * gate);   // global_atomic_add_f32
    unsafeAtomicAdd(dst + 1, v.y * gate);
    unsafeAtomicAdd(dst + 2, v.z * gate);
    unsafeAtomicAdd(dst + 3, v.w * gate);
}

// ---------------------------------------------------------------------------
extern "C" void kernel_launch(void* const* d_in, const int* in_sizes, int n_in,
                              void* d_out, int out_size, void* d_ws, size_t ws_size,
                              hipStream_t stream) {
    const float* x          = (const float*)d_in[0];
    const int*   edge_index = (const int*)  d_in[1];
    const float* edge_attr  = (const float*)d_in[2];
    const float* W_node     = (const float*)d_in[3];
    const float* b_node     = (const float*)d_in[4];
    const float* W_edge     = (const float*)d_in[5];
    const float* b_edge     = (const float*)d_in[6];
    float*       out        = (float*)d_out;
    float*       x_proj     = (float*)d_ws;   // N*128 floats (51.2 MB)

    const int N = in_sizes[0] / NODE_DIM;
    const int E = in_sizes[2] / EDGE_DIM;

    // K0: zero output
    const int n4 = (N * NODE_DIM) / 4;
    ea_zero_kernel<<<(n4 + 255) / 256, 256, 0, stream>>>((float4*)out, n4);

    // K1: node projection GEMM (fp32 WMMA)
    const int mtiles = (N + 15) / 16;
    ea_node_proj_wmma<<<mtiles, 256, 0, stream>>>(x, W_node, b_node, x_proj, N);

    // K2: fused gate + gather + scatter-add (one wave per edge)
    const long threads = (long)E * 32;
    const int  blocks  = (int)((threads + 255) / 256);
    ea_edge_scatter<<<blocks, 256, 0, stream>>>(x_proj, edge_index, edge_attr,
                                                W_edge, b_edge, out, E);
}